// E59HighwayCell_2147483648040
// MI455X (gfx1250) — compile-verified
//
#include <hip/hip_runtime.h>
#include <hip/hip_bf16.h>

typedef __attribute__((ext_vector_type(2))) float v2f;
typedef __attribute__((ext_vector_type(8))) float v8f;

// ---------------------------------------------------------------------------
// Phase 1: Y[m][n] = sum_k X[m][k] * W[n][k] + bias[n]
//   X: [M,K] row-major (x flattened [T*B, D]), W: [N,K] row-major, Y: [M,N].
//   V_WMMA_F32_16X16X4_F32 (exact fp32 matrix math on CDNA5).
//
// Workgroup tile: 128(M) x 64(N), 8 waves; wave w -> rows m0+16w..+15, all 64
// cols (4 accumulators). K blocked at 32, double-buffered in LDS:
//   Xs[2][128][36], Ws[2][64][36]  (stride 36 floats):
//     - 36*r mod 64 distinct for r=0..15  -> conflict-free ds_load_b64 frags
//     - 36*4 bytes % 16 == 0             -> aligned ds_store_b128 staging
// Pipeline: global loads for block kb+1 are issued before the wmma work on kb,
// so VMEM latency hides under 32 wmmas + LDS traffic; one barrier per block.
//
// WMMA fragment layout (wave32, 32-bit data, ISA 7.12.2):
//   A (16x4): lane L holds M=L&15, K pair = 2*(L>>4) .. +1 -> one 8B LDS read.
//   B (4x16): mirrored (N=L&15, K pair 2*(L>>4)).
//   C/D:      VGPR i -> row i + 8*(L>>4), col L&15.
// ---------------------------------------------------------------------------
__global__ __launch_bounds__(256) void gemm_xwt_bias_wmma(
    const float* __restrict__ X,     // [M,K]
    const float* __restrict__ W,     // [N,K]
    const float* __restrict__ bias,  // [N]
    float* __restrict__ Y,           // [M,N]
    int M, int N, int K)
{
  constexpr int KB  = 32;   // K block
  constexpr int LDP = 36;   // padded LDS row stride (floats)

  __shared__ float Xs[2][128][LDP];
  __shared__ float Ws[2][64][LDP];

  const int tid  = threadIdx.x;
  const int lane = tid & 31;
  const int wave = tid >> 5;

  const int n0     = blockIdx.x * 64;
  const int m_base = blockIdx.y * 128;

  // fragment addressing within a wave
  const int r  = lane & 15;             // A row / B col in tile
  const int kk = (lane >> 4) << 1;      // K sub-offset: 0 or 2

  // cooperative staging addressing: 8 col-groups of 4 floats x 32 rows
  const int colg = (tid & 7) << 2;      // 0,4,...,28
  const int row0 = tid >> 3;            // 0..31

  const float* __restrict__ Xg = X + (size_t)m_base * K;
  const float* __restrict__ Wg = W + (size_t)n0 * K;

  v8f c0 = {}, c1 = {}, c2 = {}, c3 = {};

  float4 xr[4], wr[2];

  // ---- prologue: stage K-block 0 ----
#pragma unroll
  for (int i = 0; i < 4; ++i)
    xr[i] = *(const float4*)(Xg + (size_t)(row0 + 32 * i) * K + colg);
#pragma unroll
  for (int i = 0; i < 2; ++i)
    wr[i] = *(const float4*)(Wg + (size_t)(row0 + 32 * i) * K + colg);
#pragma unroll
  for (int i = 0; i < 4; ++i)
    *(float4*)&Xs[0][row0 + 32 * i][colg] = xr[i];
#pragma unroll
  for (int i = 0; i < 2; ++i)
    *(float4*)&Ws[0][row0 + 32 * i][colg] = wr[i];
  __syncthreads();

  const int nkb = K / KB;
  for (int kb = 0; kb < nkb; ++kb) {
    const int buf = kb & 1;

    // ---- issue global loads for the next K-block (latency hides under wmma)
    if (kb + 1 < nkb) {
      const int ko = (kb + 1) * KB;
#pragma unroll
      for (int i = 0; i < 4; ++i)
        xr[i] = *(const float4*)(Xg + (size_t)(row0 + 32 * i) * K + ko + colg);
#pragma unroll
      for (int i = 0; i < 2; ++i)
        wr[i] = *(const float4*)(Wg + (size_t)(row0 + 32 * i) * K + ko + colg);
    }

    // ---- compute on current buffer: 8 k-steps x 4 wmma ----
    const float* __restrict__ xsrow = &Xs[buf][16 * wave + r][kk];
    const float* __restrict__ wsrow = &Ws[buf][r][kk];
#pragma unroll
    for (int k4 = 0; k4 < KB; k4 += 4) {
      v2f a  = *(const v2f*)(xsrow + k4);
      v2f b0 = *(const v2f*)(wsrow + k4);
      v2f b1 = *(const v2f*)(wsrow + 16 * LDP + k4);
      v2f b2 = *(const v2f*)(wsrow + 32 * LDP + k4);
      v2f b3 = *(const v2f*)(wsrow + 48 * LDP + k4);
      // (neg_a, A, neg_b, B, c_mod, C, reuse_a, reuse_b)
      c0 = __builtin_amdgcn_wmma_f32_16x16x4_f32(false, a, false, b0, (short)0, c0, false, false);
      c1 = __builtin_amdgcn_wmma_f32_16x16x4_f32(false, a, false, b1, (short)0, c1, false, false);
      c2 = __builtin_amdgcn_wmma_f32_16x16x4_f32(false, a, false, b2, (short)0, c2, false, false);
      c3 = __builtin_amdgcn_wmma_f32_16x16x4_f32(false, a, false, b3, (short)0, c3, false, false);
    }

    // ---- stage next block into the other buffer, then one barrier ----
    if (kb + 1 < nkb) {
      const int nb = buf ^ 1;
#pragma unroll
      for (int i = 0; i < 4; ++i)
        *(float4*)&Xs[nb][row0 + 32 * i][colg] = xr[i];
#pragma unroll
      for (int i = 0; i < 2; ++i)
        *(float4*)&Ws[nb][row0 + 32 * i][colg] = wr[i];
      __syncthreads();
    }
  }

  // ---- bias + store: c[i] of lane L -> Y[m0+i+8*(L>>4)][n0+16*j+(L&15)] ----
  const int m0      = m_base + 16 * wave;
  const int col     = lane & 15;
  const int row_add = (lane >> 4) << 3;       // 0 or 8
  const float bv0 = bias[n0 + col];
  const float bv1 = bias[n0 + 16 + col];
  const float bv2 = bias[n0 + 32 + col];
  const float bv3 = bias[n0 + 48 + col];

#pragma unroll
  for (int i = 0; i < 8; ++i) {
    float* yrow = Y + (size_t)(m0 + i + row_add) * N + n0 + col;
    yrow[0]  = c0[i] + bv0;
    yrow[16] = c1[i] + bv1;
    yrow[32] = c2[i] + bv2;
    yrow[48] = c3[i] + bv3;
  }
}

// ---------------------------------------------------------------------------
// Phase 2: sequential highway scan. One workgroup per batch row b (16 blocks),
// 1024 threads = one channel each; h lives in a register across all T steps.
// Per step: v = h + alpha*wx; h = v * rsqrt(mean(v^2)+eps); out = h^2*sigmoid(h)
// Reduction: wave32 shfl_xor tree + 32-entry LDS stage across the 32 waves.
// global_prefetch pulls Wx[t+2] while the current reduction is in flight.
// ---------------------------------------------------------------------------
__global__ __launch_bounds__(1024) void highway_scan(
    const float* __restrict__ Wx,     // [T,B,D] (bias already folded in)
    const float* __restrict__ h0,     // [B,D]
    const float* __restrict__ log_alpha,
    float* __restrict__ outs,         // [T,B,D]
    float* __restrict__ hout,         // [T+1,B,D]
    int T, int B, int D)
{
  const int b    = blockIdx.x;
  const int d    = threadIdx.x;
  const int lane = threadIdx.x & 31;
  const int wave = threadIdx.x >> 5;

  __shared__ float red[32];
  __shared__ float bcast;

  const float alpha = expf(log_alpha[0]);
  const float invD  = 1.0f / (float)D;
  const size_t BD   = (size_t)B * D;

  float h = h0[(size_t)b * D + d];
  hout[(size_t)b * D + d] = h;                 // h[0] = h0

  for (int t = 0; t < T; ++t) {
    const size_t idx = (size_t)t * BD + (size_t)b * D + d;
    const float wx = Wx[idx];
    if (t + 2 < T)
      __builtin_prefetch(Wx + idx + 2 * BD, 0, 0);   // global_prefetch_b8

    const float v = h + alpha * wx;

    // block-wide sum of v*v
    float s = v * v;
#pragma unroll
    for (int off = 16; off > 0; off >>= 1) s += __shfl_xor(s, off, 32);
    if (lane == 0) red[wave] = s;
    __syncthreads();
    if (wave == 0) {
      float s2 = red[lane];
#pragma unroll
      for (int off = 16; off > 0; off >>= 1) s2 += __shfl_xor(s2, off, 32);
      if (lane == 0) bcast = s2;
    }
    __syncthreads();

    const float mean = bcast * invD;
    const float rn   = rsqrtf(mean + 1e-6f);
    h = v * rn;

    const float sig = 1.0f / (1.0f + expf(-h));
    outs[idx]      = h * h * sig;              // h * silu(h)
    hout[idx + BD] = h;                        // h[t+1]
  }
}

// ---------------------------------------------------------------------------
extern "C" void kernel_launch(void* const* d_in, const int* in_sizes, int n_in,
                              void* d_out, int out_size, void* d_ws, size_t ws_size,
                              hipStream_t stream) {
  const float* x         = (const float*)d_in[0];  // [T,B,D]
  const float* h0        = (const float*)d_in[1];  // [B,D]
  const float* W         = (const float*)d_in[2];  // [D,D]
  const float* bias      = (const float*)d_in[3];  // [D]
  const float* log_alpha = (const float*)d_in[4];  // [1]

  const int D = in_sizes[3];
  const int B = in_sizes[1] / D;
  const int T = in_sizes[0] / in_sizes[1];
  const int M = T * B;

  float* Wx   = (float*)d_ws;                 // [M, D] scratch (128 MB)
  float* outs = (float*)d_out;                // [T,B,D]
  float* hout = outs + (size_t)T * B * D;     // [T+1,B,D]

  dim3 gemm_grid(D / 64, M / 128);
  gemm_xwt_bias_wmma<<<gemm_grid, 256, 0, stream>>>(x, W, bias, Wx, M, D, D);

  highway_scan<<<B, D, 0, stream>>>(Wx, h0, log_alpha, outs, hout, T, B, D);
}